// GCN_3822520893866
// MI455X (gfx1250) — compile-verified
//
#include <hip/hip_runtime.h>
#include <math.h>

// Problem constants (match reference)
#define N_NODES 50000
#define N_EDGES 800000
#define NB 4
#define NF 256
#define NH 32
#define NC 7
#define BH (NB * NH)   // 128 cols of s1 / h
#define BC (NB * NC)   // 28 cols of s2
#define MTILES (N_NODES / 16)  // 3125

typedef __attribute__((ext_vector_type(2))) float v2f;
typedef __attribute__((ext_vector_type(8))) float v8f;

// ---------------------------------------------------------------------------
// Kernel 1: CSR row_ptr from sorted COO rows (binary search, deterministic)
// ---------------------------------------------------------------------------
__global__ __launch_bounds__(256) void build_row_ptr_kernel(
    const int* __restrict__ row, int* __restrict__ rp) {
  int n = blockIdx.x * blockDim.x + threadIdx.x;
  if (n > N_NODES) return;
  if (n == N_NODES) { rp[n] = N_EDGES; return; }
  int lo = 0, hi = N_EDGES;
  while (lo < hi) {
    int mid = (lo + hi) >> 1;
    if (row[mid] < n) lo = mid + 1; else hi = mid;
  }
  rp[n] = lo;  // lower_bound(n)
}

// ---------------------------------------------------------------------------
// Kernel 2: s1[n, b*32+h] = sum_f x[n,f] * w1[b,f,h]   (f32 WMMA 16x16x4)
// One wave computes ALL 8 column tiles (128 cols) of one 16-row strip:
// A is loaded once per K-step and reused by 8 independent WMMA chains
// (8 accumulators), cutting x read-amplification 8x and giving the XDL
// pipe 8 independent D->C chains (no RAW hazard between consecutive WMMAs).
// A layout (16x4 f32): lane l, vgpr v -> M = l%16, K = 2*(l/16)+v
// B layout (4x16 f32): lane l, vgpr v -> K = v+2*(l/16), N = l%16
// C/D layout: vgpr r, lane l -> M = r+8*(l/16), N = l%16
// ---------------------------------------------------------------------------
__global__ __launch_bounds__(128) void gemm1_wmma_kernel(
    const float* __restrict__ x, const float* __restrict__ w1,
    float* __restrict__ s1) {
  const int lane = threadIdx.x & 31;
  const int mt = blockIdx.x * 4 + (threadIdx.x >> 5);
  if (mt >= MTILES) return;              // wave-uniform exit
  const int m0 = mt * 16;
  const int lm = lane & 15;
  const int lh = lane >> 4;

  const float* xrow = x + (size_t)(m0 + lm) * NF + lh * 2;

  // Per-column-tile base offset into w1[B][F][H]: col = nt*16+lm -> b=col/32,h=col%32
  int wofs[8];
#pragma unroll
  for (int nt = 0; nt < 8; ++nt) {
    int c = nt * 16 + lm;
    wofs[nt] = (c >> 5) * (NF * NH) + (c & 31);
  }

  v8f acc[8];
#pragma unroll
  for (int nt = 0; nt < 8; ++nt) acc[nt] = (v8f){};

#pragma unroll 2
  for (int k0 = 0; k0 < NF; k0 += 4) {
    const v2f a = *(const v2f*)(xrow + k0);   // 8B aligned: (lh*2+k0) even
    const int kb = k0 + lh * 2;
#pragma unroll
    for (int nt = 0; nt < 8; ++nt) {
      const float* wp = w1 + wofs[nt] + kb * NH;
      v2f bv;
      bv.x = wp[0];
      bv.y = wp[NH];
      acc[nt] = __builtin_amdgcn_wmma_f32_16x16x4_f32(
          false, a, false, bv, (short)0, acc[nt], false, false);
    }
  }

#pragma unroll
  for (int nt = 0; nt < 8; ++nt) {
    const int cb = nt * 16 + lm;
#pragma unroll
    for (int r = 0; r < 8; ++r) {
      int m = m0 + lh * 8 + r;
      s1[(size_t)m * BH + cb] = acc[nt][r];
    }
  }
}

// ---------------------------------------------------------------------------
// Kernel 3: h = relu(A @ s1). One wave per node; lane owns 4 of 128 columns.
// s1 (25.6MB) stays L2-resident -> gathers served from L2.
// ---------------------------------------------------------------------------
__global__ __launch_bounds__(128) void spmm1_relu_kernel(
    const int* __restrict__ rp, const int* __restrict__ col,
    const float* __restrict__ val, const float* __restrict__ s1,
    float* __restrict__ hout) {
  const int lane = threadIdx.x & 31;
  const int node = blockIdx.x * 4 + (threadIdx.x >> 5);
  if (node >= N_NODES) return;
  const int s = rp[node], e = rp[node + 1];
  float a0 = 0.f, a1 = 0.f, a2 = 0.f, a3 = 0.f;
  for (int i = s; i < e; ++i) {
    float v = val[i];
    int c = col[i];
    const float* src = s1 + (size_t)c * BH + lane;
    a0 = fmaf(v, src[0],  a0);
    a1 = fmaf(v, src[32], a1);
    a2 = fmaf(v, src[64], a2);
    a3 = fmaf(v, src[96], a3);
  }
  float* dst = hout + (size_t)node * BH + lane;
  dst[0]  = fmaxf(a0, 0.f);
  dst[32] = fmaxf(a1, 0.f);
  dst[64] = fmaxf(a2, 0.f);
  dst[96] = fmaxf(a3, 0.f);
}

// ---------------------------------------------------------------------------
// Kernel 4: s2[n, b*7+c] = sum_k h[n, b*32+k] * w2[b,k,c]  (WMMA, N padded to 16)
// ---------------------------------------------------------------------------
__global__ __launch_bounds__(128) void gemm2_wmma_kernel(
    const float* __restrict__ h, const float* __restrict__ w2,
    float* __restrict__ s2) {
  const int lane = threadIdx.x & 31;
  const int tile = blockIdx.x * 4 + (threadIdx.x >> 5);  // [0, 12500)
  const int mt = tile >> 2;
  const int b  = tile & 3;
  const int m0 = mt * 16;
  const int lm = lane & 15;
  const int lh = lane >> 4;

  const float* hrow = h + (size_t)(m0 + lm) * BH + b * NH + lh * 2;
  const float* wb   = w2 + b * NH * NC;  // [32, 7]
  const bool cok = (lm < NC);

  v8f acc = {};
#pragma unroll
  for (int k0 = 0; k0 < NH; k0 += 4) {
    v2f a = *(const v2f*)(hrow + k0);
    int kb = k0 + lh * 2;
    v2f bv;
    bv.x = cok ? wb[kb * NC + lm]       : 0.f;
    bv.y = cok ? wb[(kb + 1) * NC + lm] : 0.f;
    acc = __builtin_amdgcn_wmma_f32_16x16x4_f32(
        false, a, false, bv, (short)0, acc, false, false);
  }

  if (cok) {
#pragma unroll
    for (int r = 0; r < 8; ++r) {
      int m = m0 + lh * 8 + r;
      s2[(size_t)m * BC + b * NC + lm] = acc[r];
    }
  }
}

// ---------------------------------------------------------------------------
// Kernel 5: out = A @ s2, transpose to [B,C,N], log_softmax over C (7 lanes).
// One wave per node; lane = b*7+c for lane<28. d_out = logp | out (concat).
// ---------------------------------------------------------------------------
__global__ __launch_bounds__(128) void spmm2_logsoftmax_kernel(
    const int* __restrict__ rp, const int* __restrict__ col,
    const float* __restrict__ val, const float* __restrict__ s2,
    float* __restrict__ out) {
  const int lane = threadIdx.x & 31;
  const int node = blockIdx.x * 4 + (threadIdx.x >> 5);
  if (node >= N_NODES) return;
  const int li = (lane < BC) ? lane : 0;  // clamp inactive lanes
  const int s = rp[node], e = rp[node + 1];
  float acc = 0.f;
  for (int i = s; i < e; ++i) {
    float v = val[i];
    int c = col[i];
    acc = fmaf(v, s2[(size_t)c * BC + li], acc);
  }
  // log-softmax within this lane's batch group of 7 channels
  const int base = (li / NC) * NC;
  float m = -INFINITY;
#pragma unroll
  for (int j = 0; j < NC; ++j) m = fmaxf(m, __shfl(acc, base + j, 32));
  float ssum = 0.f;
#pragma unroll
  for (int j = 0; j < NC; ++j) ssum += expf(__shfl(acc, base + j, 32) - m);
  float logp = acc - m - logf(ssum);

  if (lane < BC) {
    out[(size_t)lane * N_NODES + node]        = logp;  // logp: [4,7,N] flat
    out[(size_t)(BC + lane) * N_NODES + node] = acc;   // out:  [4,7,N] flat
  }
}

// ---------------------------------------------------------------------------
extern "C" void kernel_launch(void* const* d_in, const int* in_sizes, int n_in,
                              void* d_out, int out_size, void* d_ws,
                              size_t ws_size, hipStream_t stream) {
  (void)in_sizes; (void)n_in; (void)out_size; (void)ws_size;
  const float* x    = (const float*)d_in[0];
  const int*   erow = (const int*)d_in[1];
  const int*   ecol = (const int*)d_in[2];
  const float* eval = (const float*)d_in[3];
  const float* w1   = (const float*)d_in[4];
  const float* w2   = (const float*)d_in[5];
  float* out = (float*)d_out;

  // Workspace layout (~57.0 MB total)
  char* ws = (char*)d_ws;
  size_t off = 0;
  int* rp = (int*)(ws + off);
  off += (((size_t)(N_NODES + 1) * sizeof(int)) + 255) & ~(size_t)255;
  float* s1 = (float*)(ws + off); off += (size_t)N_NODES * BH * sizeof(float);
  float* h  = (float*)(ws + off); off += (size_t)N_NODES * BH * sizeof(float);
  float* s2 = (float*)(ws + off); off += (size_t)N_NODES * BC * sizeof(float);

  build_row_ptr_kernel<<<(N_NODES + 256) / 256, 256, 0, stream>>>(erow, rp);
  gemm1_wmma_kernel<<<(MTILES + 3) / 4, 128, 0, stream>>>(x, w1, s1);
  spmm1_relu_kernel<<<N_NODES / 4, 128, 0, stream>>>(rp, ecol, eval, s1, h);
  gemm2_wmma_kernel<<<(MTILES * 4) / 4, 128, 0, stream>>>(h, w2, s2);
  spmm2_logsoftmax_kernel<<<N_NODES / 4, 128, 0, stream>>>(rp, ecol, eval, s2, out);
}